// HebbianNeuralNetwork_60825326846529
// MI455X (gfx1250) — compile-verified
//
#include <hip/hip_runtime.h>
#include <math.h>

typedef __attribute__((ext_vector_type(2))) float v2f;
typedef __attribute__((ext_vector_type(8))) float v8f;

#define INPUT   2048
#define HIDDEN  32
#define OUTPUT  512
#define BATCH   16384

// ws layout (float offsets)
#define WS_ACT        0
#define WS_XSUM       (BATCH * HIDDEN)            // 524288
#define WS_ASUM       (WS_XSUM + INPUT)
#define WS_OSUM       (WS_ASUM + HIDDEN)
#define WS_CORR1      (WS_OSUM + OUTPUT)
#define WS_CORR2      (WS_CORR1 + INPUT * HIDDEN)
#define WS_TOTAL      (WS_CORR2 + HIDDEN * OUTPUT)
#define WS_ACCUM_N    (WS_TOTAL - WS_XSUM)

#define K1SPLIT 4
#define K2SPLIT 16

// k_act LDS staging: K-chunk of w1, padded so B-operand ds_load_b64 is bank-conflict-free
#define KC   256
#define NKC  (INPUT / KC)     // 8
#define KCP  (KC + 4)         // pad: lane stride = 4 banks -> 32 b64 reads hit all 64 banks once

static __device__ __forceinline__ v8f wmma_f32(v2f a, v2f b, v8f c) {
    return __builtin_amdgcn_wmma_f32_16x16x4_f32(false, a, false, b, (short)0, c,
                                                 false, false);
}

// CDNA5 async global->LDS copy (16B per lane), tracked by ASYNCcnt.
static __device__ __forceinline__ void async_cp16(const float* g, uint32_t lds_addr) {
    asm volatile("global_load_async_to_lds_b128 %0, %1, off"
                 :: "v"(lds_addr), "v"(g) : "memory");
}
static __device__ __forceinline__ void wait_async0() {
    asm volatile("s_wait_asynccnt 0x0" ::: "memory");
}

__global__ void k_zero(float* __restrict__ p, int n) {
    int i = blockIdx.x * blockDim.x + threadIdx.x;
    if (i < n) p[i] = 0.0f;
}

// act = relu(x @ w1^T)  [BATCH, HIDDEN]; also act_sum += column sums.
// w1 is staged through LDS with double-buffered async DMA; B reads become ds_load_b64.
__global__ __launch_bounds__(256) void k_act(const float* __restrict__ x,
                                             const float* __restrict__ w1,
                                             float* __restrict__ act,
                                             float* __restrict__ act_sum) {
    __shared__ float sw1[2][HIDDEN * KCP];

    const int tid  = threadIdx.x;
    const int lane = tid & 31;
    const int wave = tid >> 5;
    const int m0   = (blockIdx.x * 8 + wave) * 16;
    const int lo   = lane & 15;
    const int hi   = lane >> 4;

    // Stage one K-chunk of w1 (32 rows x KC floats) into LDS buffer `buf`.
    // 2048 float4 elements, 256 threads -> 8 async b128 ops per thread.
    auto stage = [&](int buf, int kc0) {
#pragma unroll
        for (int j = 0; j < (HIDDEN * KC / 4) / 256; ++j) {
            int e  = j * 256 + tid;
            int r  = e >> 6;          // KC/4 = 64 float4 per row
            int c4 = e & 63;
            const float* g = w1 + (size_t)r * INPUT + kc0 + c4 * 4;
            uint32_t lds_addr =
                (uint32_t)(uintptr_t)(&sw1[buf][r * KCP + c4 * 4]);
            async_cp16(g, lds_addr);
        }
    };

    stage(0, 0);

    const float* xrow = x + (size_t)(m0 + lo) * INPUT + 2 * hi;

    v8f c0 = {0.f,0.f,0.f,0.f,0.f,0.f,0.f,0.f};
    v8f c1 = {0.f,0.f,0.f,0.f,0.f,0.f,0.f,0.f};

    for (int kc = 0; kc < NKC; ++kc) {
        const int buf = kc & 1;
        wait_async0();            // this wave's async copies done (incl. buffer `buf`)
        __syncthreads();          // all waves' copies visible in LDS
        if (kc + 1 < NKC) stage(buf ^ 1, (kc + 1) * KC);

        const float* bw = &sw1[buf][0];
#pragma unroll 8
        for (int k = 0; k < KC; k += 4) {
            v2f a  = *(const v2f*)(xrow + kc * KC + k);
            v2f b0 = *(const v2f*)(bw + (size_t)lo * KCP + k + 2 * hi);
            v2f b1 = *(const v2f*)(bw + (size_t)(lo + 16) * KCP + k + 2 * hi);
            c0 = wmma_f32(a, b0, c0);
            c1 = wmma_f32(a, b1, c1);
        }
        __syncthreads();          // everyone done reading `buf` before it is re-staged
    }

    float s0 = 0.f, s1 = 0.f;
#pragma unroll
    for (int j = 0; j < 8; ++j) {
        float v0 = c0[j] > 0.f ? c0[j] : 0.f;
        float v1 = c1[j] > 0.f ? c1[j] : 0.f;
        int row = m0 + j + 8 * hi;
        act[(size_t)row * HIDDEN + lo]      = v0;
        act[(size_t)row * HIDDEN + lo + 16] = v1;
        s0 += v0;
        s1 += v1;
    }
    atomicAdd(&act_sum[lo],      s0);
    atomicAdd(&act_sum[lo + 16], s1);
}

// out = sigmoid(act @ w2^T) - 0.4  [BATCH, OUTPUT]
__global__ __launch_bounds__(256) void k_out(const float* __restrict__ act,
                                             const float* __restrict__ w2,
                                             float* __restrict__ out) {
    const int lane = threadIdx.x & 31;
    const int wave = threadIdx.x >> 5;
    const int tile = blockIdx.x * 8 + wave;      // 0..32767
    const int mt   = tile >> 5;                  // 0..1023
    const int nt   = tile & 31;                  // 0..31
    const int m0 = mt * 16, n0 = nt * 16;
    const int lo = lane & 15, hi = lane >> 4;

    const float* arow = act + (size_t)(m0 + lo) * HIDDEN + 2 * hi;
    const float* wrow = w2  + (size_t)(n0 + lo) * HIDDEN + 2 * hi;

    v8f c = {0.f,0.f,0.f,0.f,0.f,0.f,0.f,0.f};
#pragma unroll
    for (int k = 0; k < HIDDEN; k += 4) {
        v2f a = *(const v2f*)(arow + k);
        v2f b = *(const v2f*)(wrow + k);
        c = wmma_f32(a, b, c);
    }
#pragma unroll
    for (int j = 0; j < 8; ++j) {
        float v = 1.0f / (1.0f + __expf(-c[j])) - 0.4f;
        out[(size_t)(m0 + j + 8 * hi) * OUTPUT + n0 + lo] = v;
    }
}

// corr1 = x^T @ act  [INPUT, HIDDEN] (K = BATCH, split), also x_sum.
__global__ __launch_bounds__(256) void k_corr1(const float* __restrict__ x,
                                               const float* __restrict__ act,
                                               float* __restrict__ corr1,
                                               float* __restrict__ x_sum) {
    const int lane = threadIdx.x & 31;
    const int wave = threadIdx.x >> 5;
    const int it   = blockIdx.x / K1SPLIT;                 // 0..127
    const int ks   = (blockIdx.x % K1SPLIT) * 8 + wave;    // 0..31
    const int i0   = it * 16;
    const int lo = lane & 15, hi = lane >> 4;
    const int kChunk = BATCH / (K1SPLIT * 8);              // 512
    const int kBeg   = ks * kChunk;

    const float* xcol = x   + (size_t)kBeg * INPUT + i0 + lo;
    const float* acol = act + (size_t)kBeg * HIDDEN + lo;

    v8f c0 = {0.f,0.f,0.f,0.f,0.f,0.f,0.f,0.f};
    v8f c1 = {0.f,0.f,0.f,0.f,0.f,0.f,0.f,0.f};
    float xs = 0.f;
    for (int k = 0; k < kChunk; k += 4) {
        int b = k + 2 * hi;
        float a0 = xcol[(size_t)b * INPUT];
        float a1 = xcol[(size_t)(b + 1) * INPUT];
        v2f a = {a0, a1};
        xs += a0 + a1;
        v2f b0 = { acol[(size_t)b * HIDDEN],      acol[(size_t)(b + 1) * HIDDEN] };
        v2f b1 = { acol[(size_t)b * HIDDEN + 16], acol[(size_t)(b + 1) * HIDDEN + 16] };
        c0 = wmma_f32(a, b0, c0);
        c1 = wmma_f32(a, b1, c1);
    }
#pragma unroll
    for (int j = 0; j < 8; ++j) {
        int row = i0 + j + 8 * hi;
        atomicAdd(&corr1[(size_t)row * HIDDEN + lo],      c0[j]);
        atomicAdd(&corr1[(size_t)row * HIDDEN + lo + 16], c1[j]);
    }
    atomicAdd(&x_sum[i0 + lo], xs);
}

// corr2 = act^T @ out  [HIDDEN, OUTPUT] (K = BATCH, split), also out_sum.
__global__ __launch_bounds__(256) void k_corr2(const float* __restrict__ act,
                                               const float* __restrict__ out,
                                               float* __restrict__ corr2,
                                               float* __restrict__ out_sum) {
    const int lane = threadIdx.x & 31;
    const int wave = threadIdx.x >> 5;
    const int nt = blockIdx.x / K2SPLIT;                   // 0..31
    const int ks = (blockIdx.x % K2SPLIT) * 8 + wave;      // 0..127
    const int n0 = nt * 16;
    const int lo = lane & 15, hi = lane >> 4;
    const int kChunk = BATCH / (K2SPLIT * 8);              // 128
    const int kBeg   = ks * kChunk;

    const float* acol = act + (size_t)kBeg * HIDDEN + lo;
    const float* ocol = out + (size_t)kBeg * OUTPUT + n0 + lo;

    v8f c0 = {0.f,0.f,0.f,0.f,0.f,0.f,0.f,0.f};
    v8f c1 = {0.f,0.f,0.f,0.f,0.f,0.f,0.f,0.f};
    float os = 0.f;
    for (int k = 0; k < kChunk; k += 4) {
        int b = k + 2 * hi;
        v2f a0 = { acol[(size_t)b * HIDDEN],      acol[(size_t)(b + 1) * HIDDEN] };
        v2f a1 = { acol[(size_t)b * HIDDEN + 16], acol[(size_t)(b + 1) * HIDDEN + 16] };
        float bv0 = ocol[(size_t)b * OUTPUT];
        float bv1 = ocol[(size_t)(b + 1) * OUTPUT];
        v2f bb = {bv0, bv1};
        os += bv0 + bv1;
        c0 = wmma_f32(a0, bb, c0);
        c1 = wmma_f32(a1, bb, c1);
    }
#pragma unroll
    for (int j = 0; j < 8; ++j) {
        int h = j + 8 * hi;
        atomicAdd(&corr2[(size_t)h * OUTPUT + n0 + lo],        c0[j]);
        atomicAdd(&corr2[(size_t)(h + 16) * OUTPUT + n0 + lo], c1[j]);
    }
    atomicAdd(&out_sum[n0 + lo], os);
}

// w1_new = rownorm(w1 + dw1T^T); one block per hidden row h.
__global__ __launch_bounds__(256) void k_w1(const float* __restrict__ w1,
                                            const float* __restrict__ heb,
                                            const float* __restrict__ corr1,
                                            const float* __restrict__ x_sum,
                                            const float* __restrict__ act_sum,
                                            float* __restrict__ w1_out) {
    const int h   = blockIdx.x;
    const int tid = threadIdx.x;
    const float asum_h = act_sum[h];
    float vals[INPUT / 256];
    float ss = 0.f;
#pragma unroll
    for (int r = 0; r < INPUT / 256; ++r) {
        int i = tid + r * 256;
        float4 cc = *(const float4*)(heb + ((size_t)i * HIDDEN + h) * 4);
        float dw = (float)BATCH * cc.w + cc.x * corr1[(size_t)i * HIDDEN + h]
                 + cc.y * x_sum[i] + cc.z * asum_h;
        float v = w1[(size_t)h * INPUT + i] + dw;
        vals[r] = v;
        ss += v * v;
    }
    __shared__ float red[8];
    for (int off = 16; off > 0; off >>= 1) ss += __shfl_xor(ss, off, 32);
    if ((tid & 31) == 0) red[tid >> 5] = ss;
    __syncthreads();
    float total = 0.f;
#pragma unroll
    for (int w = 0; w < 8; ++w) total += red[w];
    float rn = 1.0f / sqrtf(total);
#pragma unroll
    for (int r = 0; r < INPUT / 256; ++r) {
        int i = tid + r * 256;
        w1_out[(size_t)h * INPUT + i] = vals[r] * rn;
    }
}

// w2_new = rownorm(w2 + dw2T^T); one wave per output row (32 lanes == HIDDEN).
__global__ __launch_bounds__(256) void k_w2(const float* __restrict__ w2,
                                            const float* __restrict__ heb2,
                                            const float* __restrict__ corr2,
                                            const float* __restrict__ act_sum,
                                            const float* __restrict__ out_sum,
                                            float* __restrict__ w2_out) {
    const int lane = threadIdx.x & 31;   // hidden index
    const int wave = threadIdx.x >> 5;
    const int o = blockIdx.x * 8 + wave;
    float4 cc = *(const float4*)(heb2 + ((size_t)lane * OUTPUT + o) * 4);
    float dw = (float)BATCH * cc.w + cc.x * corr2[(size_t)lane * OUTPUT + o]
             + cc.y * act_sum[lane] + cc.z * out_sum[o];
    float v = w2[(size_t)o * HIDDEN + lane] + dw;
    float ss = v * v;
    for (int off = 16; off > 0; off >>= 1) ss += __shfl_xor(ss, off, 32);
    w2_out[(size_t)o * HIDDEN + lane] = v / sqrtf(ss);
}

extern "C" void kernel_launch(void* const* d_in, const int* in_sizes, int n_in,
                              void* d_out, int out_size, void* d_ws, size_t ws_size,
                              hipStream_t stream) {
    const float* x   = (const float*)d_in[0];
    const float* w1  = (const float*)d_in[1];
    const float* w2  = (const float*)d_in[2];
    const float* heb = (const float*)d_in[3];

    float* out    = (float*)d_out;                         // [BATCH, OUTPUT]
    float* w1_out = out + (size_t)BATCH * OUTPUT;          // [HIDDEN, INPUT]
    float* w2_out = w1_out + (size_t)HIDDEN * INPUT;       // [OUTPUT, HIDDEN]

    float* ws      = (float*)d_ws;
    float* act     = ws + WS_ACT;
    float* x_sum   = ws + WS_XSUM;
    float* act_sum = ws + WS_ASUM;
    float* out_sum = ws + WS_OSUM;
    float* corr1   = ws + WS_CORR1;
    float* corr2   = ws + WS_CORR2;

    k_zero<<<(WS_ACCUM_N + 255) / 256, 256, 0, stream>>>(ws + WS_XSUM, WS_ACCUM_N);
    k_act<<<BATCH / 128, 256, 0, stream>>>(x, w1, act, act_sum);
    k_out<<<(BATCH / 16) * (OUTPUT / 16) / 8, 256, 0, stream>>>(act, w2, out);
    k_corr1<<<(INPUT / 16) * K1SPLIT, 256, 0, stream>>>(x, act, corr1, x_sum);
    k_corr2<<<(OUTPUT / 16) * K2SPLIT, 256, 0, stream>>>(act, out, corr2, out_sum);
    k_w1<<<HIDDEN, 256, 0, stream>>>(w1, heb, corr1, x_sum, act_sum, w1_out);
    k_w2<<<OUTPUT / 8, 256, 0, stream>>>(w2, heb + (size_t)HIDDEN * INPUT * 4,
                                         corr2, act_sum, out_sum, w2_out);
}